// TemporalRankingLoss_523986010373
// MI455X (gfx1250) — compile-verified
//
#include <hip/hip_runtime.h>
#include <math.h>

// TemporalRankingLoss on gfx1250 (MI455X, wave32).
//
// Stage 1 (bandwidth-dominant, 32 MB @ 23.3 TB/s ~= 1.4 us): d_high via
//   V_WMMA_F32_16X16X4_F32 with B = ones -> D[m][*] = rowsum of squared
//   diffs. One workgroup = 4 waves per 16-edge tile; each wave reduces a
//   dh/4 slice of the K dimension into its own dual WMMA accumulator chains
//   (b128 loads, 4 dims/lane/iter), then a small LDS reduction combines the
//   4 partial rowsums. 4x the resident waves (~2048) hides the per-wave
//   load->wait->consume latency; B=ones makes K-placement arbitrary, only
//   the lane->row mapping (lane&15) matters.
// Stage 2: d_low (DE=2), group boundaries from sorted segment_ids, zero accums.
// Stage 3: group-local pairwise scan (~E*avg_group ops instead of E^2),
//   wave32 shuffle reduction + one atomic per wave.
// Stage 4: finalize scalar.

typedef __attribute__((ext_vector_type(2))) float v2f;
typedef __attribute__((ext_vector_type(8))) float v8f;

#define KWAVES 4   // waves per 16-edge tile (K-dimension split); dh % (8*KWAVES) == 0

__global__ __launch_bounds__(32 * KWAVES)
void dhigh_wmma_kernel(const float* __restrict__ to_,
                       const float* __restrict__ from_,
                       float* __restrict__ dhigh, int dh) {
  __shared__ float red[KWAVES][16];

  const int tid  = threadIdx.x;
  const int wave = tid >> 5;               // 0..KWAVES-1
  const int lane = tid & 31;               // wave32 lane; EXEC all-ones in WMMA loop
  const int m    = lane & 15;              // row within 16-edge tile (A-matrix M)
  const int kh   = (lane >> 4) << 2;       // lanes 0-15: dims +0..3 ; 16-31: +4..7
  const int kspan = dh / KWAVES;           // 128 dims per wave
  const long long row = ((long long)blockIdx.x * 16 + m) * (long long)dh;
  const float* tp = to_ + row + (long long)wave * kspan + kh;
  const float* fp = from_ + row + (long long)wave * kspan + kh;

  v8f acc0 = {};                           // two independent C/D chains (8 VGPRs each)
  v8f acc1 = {};
  v2f onesv; onesv[0] = 1.0f; onesv[1] = 1.0f;   // B = all ones (4x16)

  for (int k = 0; k < kspan; k += 8) {
    // 16B-aligned: row multiple of dh floats, wave*kspan + k + kh multiple of 4
    float4 t = *(const float4*)(tp + k);
    float4 f = *(const float4*)(fp + k);
    float d0 = t.x - f.x, d1 = t.y - f.y, d2 = t.z - f.z, d3 = t.w - f.w;
    v2f a0; a0[0] = d0 * d0; a0[1] = d1 * d1;
    v2f a1; a1[0] = d2 * d2; a1[1] = d3 * d3;
    // D = A x ones + C : every column of row m accumulates partial rowsum(m)
    acc0 = __builtin_amdgcn_wmma_f32_16x16x4_f32(
        false, a0, false, onesv, (short)0, acc0, false, false);
    acc1 = __builtin_amdgcn_wmma_f32_16x16x4_f32(
        false, a1, false, onesv, (short)0, acc1, false, false);
  }
  v8f acc = acc0 + acc1;

  // C/D layout: VGPR j on lanes 0-15 -> M=j ; lanes 16-31 -> M=j+8.
  // Lane 0 of each wave stores rows 0..7, lane 16 stores rows 8..15.
  if (m == 0) {
    const int half = lane >> 4;            // 0 or 1
    float* rp = &red[wave][half * 8];
    rp[0] = acc[0]; rp[1] = acc[1]; rp[2] = acc[2]; rp[3] = acc[3];
    rp[4] = acc[4]; rp[5] = acc[5]; rp[6] = acc[6]; rp[7] = acc[7];
  }
  __syncthreads();

  // cross-wave reduction: 16 threads, one per edge row
  if (tid < 16) {
    float s = 0.0f;
    #pragma unroll
    for (int w = 0; w < KWAVES; ++w) s += red[w][tid];
    dhigh[(long long)blockIdx.x * 16 + tid] = sqrtf(s);
  }
}

__global__ __launch_bounds__(256)
void prep_kernel(const float* __restrict__ embt, const float* __restrict__ embf,
                 const int* __restrict__ seg,
                 float* __restrict__ dlow, int* __restrict__ gstart,
                 int* __restrict__ gend,
                 float* __restrict__ sumAcc, unsigned int* __restrict__ cntAcc,
                 int E, int de) {
  int e = blockIdx.x * blockDim.x + threadIdx.x;
  if (e == 0) { *sumAcc = 0.0f; *cntAcc = 0u; }   // re-init accumulators every call
  if (e >= E) return;
  float s = 0.0f;
  for (int j = 0; j < de; ++j) {
    float d = embt[e * de + j] - embf[e * de + j];
    s += d * d;
  }
  dlow[e] = sqrtf(s);
  // segment_ids is sorted -> each group occupies a contiguous [start,end) span
  int g = seg[e];
  if (e == 0 || seg[e - 1] != g) gstart[g] = e;
  if (e == E - 1 || seg[e + 1] != g) gend[g] = e + 1;
}

__global__ __launch_bounds__(256)
void pair_kernel(const float* __restrict__ dhigh, const float* __restrict__ dlow,
                 const int* __restrict__ seg, const unsigned char* __restrict__ tmp,
                 const int* __restrict__ gstart, const int* __restrict__ gend,
                 float* __restrict__ sumAcc, unsigned int* __restrict__ cntAcc,
                 int E) {
  int i = blockIdx.x * blockDim.x + threadIdx.x;
  float lsum = 0.0f;
  unsigned int lcnt = 0u;
  if (i < E && tmp[i]) {
    int g = seg[i];
    float dhi = dhigh[i];
    float dli = dlow[i];
    int k1 = gend[g];
    for (int k = gstart[g]; k < k1; ++k) {
      if (!tmp[k]) continue;
      float dlk = dlow[k];
      if (dhi < dhigh[k] && dli >= dlk) {   // rank_lt & violation (i==k excluded by strict <)
        lsum += dli - dlk;
        lcnt += 1u;
      }
    }
  }
  // wave32 reduction, then one atomic per wave
  #pragma unroll
  for (int off = 16; off > 0; off >>= 1) {
    lsum += __shfl_down(lsum, off, 32);
    lcnt += __shfl_down(lcnt, off, 32);
  }
  if ((threadIdx.x & 31) == 0) {
    atomicAdd(sumAcc, lsum);
    atomicAdd(cntAcc, lcnt);
  }
}

__global__ void finalize_kernel(const float* __restrict__ sumAcc,
                                const unsigned int* __restrict__ cntAcc,
                                float* __restrict__ out) {
  out[0] = (*cntAcc > 0u) ? (*sumAcc / (float)(*cntAcc)) : 0.0f;
}

extern "C" void kernel_launch(void* const* d_in, const int* in_sizes, int n_in,
                              void* d_out, int out_size, void* d_ws, size_t ws_size,
                              hipStream_t stream) {
  // setup_inputs() order:
  //   0 edge_to [E,DH] f32, 1 edge_from [E,DH] f32,
  //   2 embedding_to [E,DE] f32, 3 embedding_from [E,DE] f32,
  //   4 segment_ids [E] i32 (sorted), 5 is_temporal [E] bool (1 byte each)
  const float*         edge_to   = (const float*)d_in[0];
  const float*         edge_from = (const float*)d_in[1];
  const float*         emb_to    = (const float*)d_in[2];
  const float*         emb_from  = (const float*)d_in[3];
  const int*           seg       = (const int*)d_in[4];
  const unsigned char* tmp       = (const unsigned char*)d_in[5];

  const int E  = in_sizes[4];          // 8192
  const int DH = in_sizes[0] / E;      // 512 (multiple of 8*KWAVES)
  const int DE = in_sizes[2] / E;      // 2
  const int G  = 256;                  // segment ids drawn from [0, 256)

  // workspace layout (~68 KB)
  float*        dhigh  = (float*)d_ws;           // E
  float*        dlow   = dhigh + E;              // E
  int*          gstart = (int*)(dlow + E);       // G
  int*          gend   = gstart + G;             // G
  float*        sumAcc = (float*)(gend + G);     // 1
  unsigned int* cntAcc = (unsigned int*)(sumAcc + 1); // 1
  float*        out    = (float*)d_out;

  dhigh_wmma_kernel<<<E / 16, 32 * KWAVES, 0, stream>>>(edge_to, edge_from, dhigh, DH);
  prep_kernel<<<(E + 255) / 256, 256, 0, stream>>>(emb_to, emb_from, seg, dlow,
                                                   gstart, gend, sumAcc, cntAcc, E, DE);
  pair_kernel<<<(E + 255) / 256, 256, 0, stream>>>(dhigh, dlow, seg, tmp,
                                                   gstart, gend, sumAcc, cntAcc, E);
  finalize_kernel<<<1, 1, 0, stream>>>(sumAcc, cntAcc, out);
}